// GraphConv_5609227288949
// MI455X (gfx1250) — compile-verified
//
#include <hip/hip_runtime.h>

typedef float v2f __attribute__((ext_vector_type(2)));
typedef float v8f __attribute__((ext_vector_type(8)));

#define DIN  256
#define DOUT 256
#define KNB  16

#define LDS_PAD   260   // 256 + 4 -> conflict-free ds_load_b64 fragment reads
#define HALF_ROWS 128   // stage W in two 128-row halves (130 KB LDS each)
#define WPB       8     // waves per block (256 threads)

// ---------------------------------------------------------------------------
// Kernel 1: supports[n][d] = sum_k A[n][k] * W[d][k] + bias[d]
// Block = 8 waves = 128 output rows. W staged in LDS (two halves), B frags
// read via LDS (bank-conflict-free), A frags from global (WGP$-resident).
// Each wave: 16x256 C tile in 128 VGPRs, 1024 V_WMMA_F32_16X16X4_F32.
// ---------------------------------------------------------------------------
__global__ __launch_bounds__(256) void gemm_f32wmma_lds(
    const float* __restrict__ A, const float* __restrict__ W,
    const float* __restrict__ bias, float* __restrict__ out, int nrows) {
  __shared__ float sW[HALF_ROWS * LDS_PAD];     // 133,120 B

  const int tid    = threadIdx.x;
  const int waveId = tid >> 5;
  const int lane   = tid & 31;
  const int m0     = (blockIdx.x * WPB + waveId) << 4;   // 16 rows per wave
  const int lo     = lane & 15;
  const int hi     = lane >> 4;                 // 0 or 1
  const int khalf  = hi << 1;                   // 0 or 2

  // Clamp A row for out-of-range waves (stores are predicated; barriers must
  // be executed by the whole block, so no early return).
  int arowIdx = m0 + lo;
  if (arowIdx >= nrows) arowIdx = nrows - 1;
  const float* arow = A + (size_t)arowIdx * DIN;

  v8f c[16] = {};                               // 16 n-tiles of 16x16 f32 C

  for (int half = 0; half < 2; ++half) {
    __syncthreads();
    // Cooperative stage of 128 W rows into padded LDS (float4 chunks).
    for (int i = tid; i < HALF_ROWS * (DIN / 4); i += 256) {
      const int r  = i >> 6;                    // / (DIN/4) = /64: row 0..127
      const int c4 = (i & 63) << 2;             // col in steps of 4
      const float4 v =
          *(const float4*)(W + (size_t)(half * HALF_ROWS + r) * DIN + c4);
      float* dst = sW + r * LDS_PAD + c4;
      dst[0] = v.x; dst[1] = v.y; dst[2] = v.z; dst[3] = v.w;
    }
    __syncthreads();

    for (int k = 0; k < DIN; k += 4) {
      // A fragment: A[row][k+khalf .. +1]  (8-byte load)
      v2f a = *(const v2f*)(arow + k + khalf);
#pragma unroll
      for (int tt = 0; tt < 8; ++tt) {
        const int t    = half * 8 + tt;
        const int nloc = (tt << 4) + lo;        // row within staged half
        // B = W^T: lane holds B[k'][n] = W[n][k'], from LDS, banks all distinct
        v2f bf = *(const v2f*)(sW + nloc * LDS_PAD + k + khalf);
        // (neg_a, A, neg_b, B, c_mod, C, reuse_a, reuse_b)
        c[t] = __builtin_amdgcn_wmma_f32_16x16x4_f32(
            false, a, false, bf, (short)0, c[t], false, false);
      }
    }
  }

  // Epilogue: add bias, store. C VGPR r holds row (r + 8*hi), col lo of tile.
  const int r0 = m0 + (hi << 3);
#pragma unroll
  for (int t = 0; t < 16; ++t) {
    const int n    = (t << 4) + lo;
    const float bv = bias[n];
#pragma unroll
    for (int r = 0; r < 8; ++r) {
      if (r0 + r < nrows) out[(size_t)(r0 + r) * DOUT + n] = c[t][r] + bv;
    }
  }
}

// ---------------------------------------------------------------------------
// Kernel 2: per source row s: tmp[w] = src + leaky(sum_k mask*neigh + src)
// One wave per source row; lane handles 8 contiguous columns (2x float4).
// Reads original supports from d_out (L2-resident: 102.4 MB < 192 MB L2),
// writes compacted rows to d_ws -- no in-place hazard with the gathers.
// ---------------------------------------------------------------------------
__global__ __launch_bounds__(256) void aggregate(
    const float* __restrict__ sup, const int* __restrict__ neighs,
    const float* __restrict__ mask, const int* __restrict__ src_idx,
    float* __restrict__ tmp, int S) {
  const int w = blockIdx.x * (blockDim.x >> 5) + (threadIdx.x >> 5);
  if (w >= S) return;
  const int lane = threadIdx.x & 31;
  const int col  = lane << 3;

  float acc[8] = {0.f, 0.f, 0.f, 0.f, 0.f, 0.f, 0.f, 0.f};
#pragma unroll
  for (int k = 0; k < KNB; ++k) {
    const int   idx = neighs[w * KNB + k];
    const float m   = mask[w * KNB + k];
    const float* nrow = sup + (size_t)idx * DOUT + col;
    const float4 x0 = *(const float4*)(nrow);
    const float4 x1 = *(const float4*)(nrow + 4);
    acc[0] = fmaf(m, x0.x, acc[0]);
    acc[1] = fmaf(m, x0.y, acc[1]);
    acc[2] = fmaf(m, x0.z, acc[2]);
    acc[3] = fmaf(m, x0.w, acc[3]);
    acc[4] = fmaf(m, x1.x, acc[4]);
    acc[5] = fmaf(m, x1.y, acc[5]);
    acc[6] = fmaf(m, x1.z, acc[6]);
    acc[7] = fmaf(m, x1.w, acc[7]);
  }

  const int s = src_idx[w];
  const float* srow = sup + (size_t)s * DOUT + col;
  const float4 s0 = *(const float4*)(srow);
  const float4 s1 = *(const float4*)(srow + 4);
  float sv[8] = {s0.x, s0.y, s0.z, s0.w, s1.x, s1.y, s1.z, s1.w};

  float r[8];
#pragma unroll
  for (int j = 0; j < 8; ++j) {
    float v = acc[j] + sv[j];                 // pre-activation
    v = (v >= 0.f) ? v : 0.2f * v;            // LeakyReLU(0.2)
    r[j] = sv[j] + v;                         // scatter-add result
  }
  float4 o0 = {r[0], r[1], r[2], r[3]};
  float4 o1 = {r[4], r[5], r[6], r[7]};
  *(float4*)(tmp + (size_t)w * DOUT + col)     = o0;
  *(float4*)(tmp + (size_t)w * DOUT + col + 4) = o1;
}

// ---------------------------------------------------------------------------
// Kernel 3: scatter compacted rows back: out[src_idx[w]][:] = tmp[w][:]
// ---------------------------------------------------------------------------
__global__ __launch_bounds__(256) void scatter_rows(
    const float* __restrict__ tmp, const int* __restrict__ src_idx,
    float* __restrict__ out, int S) {
  const size_t i = (size_t)blockIdx.x * blockDim.x + threadIdx.x;
  if (i >= (size_t)S * DOUT) return;
  const int w = (int)(i >> 8);                 // / 256
  const int c = (int)(i & 255);
  out[(size_t)src_idx[w] * DOUT + c] = tmp[i];
}

// ---------------------------------------------------------------------------
extern "C" void kernel_launch(void* const* d_in, const int* in_sizes, int n_in,
                              void* d_out, int out_size, void* d_ws,
                              size_t ws_size, hipStream_t stream) {
  const float* word_vectors = (const float*)d_in[0];
  const int*   src_idx      = (const int*)d_in[1];
  const int*   neighs       = (const int*)d_in[2];
  const float* mask         = (const float*)d_in[3];
  const float* W            = (const float*)d_in[4];
  const float* bias         = (const float*)d_in[5];
  float* out = (float*)d_out;
  float* tmp = (float*)d_ws;                   // needs S*256*4 = 51.2 MB

  const int N = in_sizes[0] / DIN;             // 100000
  const int S = in_sizes[1];                   // 50000

  // Phase 1: supports = A @ W^T + b  (fp32 WMMA, W staged in LDS) -> d_out
  const int rowsPerBlock = WPB * 16;           // 128
  gemm_f32wmma_lds<<<(N + rowsPerBlock - 1) / rowsPerBlock, 256, 0, stream>>>(
      word_vectors, W, bias, out, N);

  // Phase 2: masked neighbor aggregation + residual + LeakyReLU -> d_ws
  aggregate<<<(S + WPB - 1) / WPB, 32 * WPB, 0, stream>>>(out, neighs, mask,
                                                          src_idx, tmp, S);

  // Phase 3: scatter results back into supports rows
  const size_t elems = (size_t)S * DOUT;
  scatter_rows<<<(unsigned)((elems + 255) / 256), 256, 0, stream>>>(tmp, src_idx,
                                                                    out, S);
}